// Eucl_Mat_47253230191386
// MI455X (gfx1250) — compile-verified
//
#include <hip/hip_runtime.h>

typedef __attribute__((ext_vector_type(2))) float v2f;
typedef __attribute__((ext_vector_type(8))) float v8f;

#define EM_N 8192
#define EM_D 512

// ---------------------------------------------------------------------------
// Kernel 1: per-row squared norms -> d_ws  (one wave32 per row)
// ---------------------------------------------------------------------------
__global__ __launch_bounds__(256)
void em_sqnorm(const float* __restrict__ A, float* __restrict__ sq) {
  const int lane = threadIdx.x & 31;
  const int wid  = threadIdx.x >> 5;            // 8 waves per block
  const int row  = blockIdx.x * 8 + wid;
  const float* __restrict__ r = A + (size_t)row * EM_D;
  float s = 0.0f;
#pragma unroll
  for (int c = lane; c < EM_D; c += 32) {
    float v = r[c];
    s = fmaf(v, v, s);
  }
#pragma unroll
  for (int off = 16; off > 0; off >>= 1)
    s += __shfl_xor(s, off, 32);                // wave32 reduction
  if (lane == 0) sq[row] = s;
}

// ---------------------------------------------------------------------------
// Kernel 2: Gram via V_WMMA_F32_16X16X4_F32, fused distance epilogue.
// Block = 256 threads = 8 waves (4 along M x 2 along N).
// Block tile = 128x128, wave tile = 32x64 = 2x4 WMMA 16x16 tiles.
// A fragment (16x4 f32): lane (h*16 + l) holds row M=l, K = 2h + {0,1}.
// B fragment (4x16 f32): lane (h*16 + l) holds col N=l, K = 2h + {0,1}.
// For the symmetric Gram (B = A^T tile) both are the same float2 load pattern.
// ---------------------------------------------------------------------------
__global__ __launch_bounds__(256, 2)
void em_dist(const float* __restrict__ A, const float* __restrict__ sq,
             float* __restrict__ out) {
  const int lane = threadIdx.x & 31;
  const int wid  = threadIdx.x >> 5;   // 0..7
  const int wm   = wid & 3;            // wave row in block   (0..3)
  const int wn   = wid >> 2;           // wave col in block   (0..1)
  const int l16  = lane & 15;
  const int half = lane >> 4;          // 0: K={0,1}, 1: K={2,3}

  const int m0 = blockIdx.y * 128 + wm * 32;   // wave M base
  const int n0 = blockIdx.x * 128 + wn * 64;   // wave N base

  // Per-lane fragment base pointers (8B-aligned float2 loads, L2-resident).
  const float* aPtr[2];
  aPtr[0] = A + (size_t)(m0 +      l16) * EM_D + half * 2;
  aPtr[1] = A + (size_t)(m0 + 16 + l16) * EM_D + half * 2;
  const float* bPtr[4];
#pragma unroll
  for (int j = 0; j < 4; ++j)
    bPtr[j] = A + (size_t)(n0 + j * 16 + l16) * EM_D + half * 2;

  const v8f vzero = {};
  v8f acc[2][4];
#pragma unroll
  for (int i = 0; i < 2; ++i)
#pragma unroll
    for (int j = 0; j < 4; ++j)
      acc[i][j] = vzero;

  // K loop: 128 iterations of 6 float2 loads + 8 f32 WMMAs.
#pragma unroll 2
  for (int k0 = 0; k0 < EM_D; k0 += 4) {
    v2f af[2], bf[4];
    af[0] = *(const v2f*)(aPtr[0] + k0);
    af[1] = *(const v2f*)(aPtr[1] + k0);
#pragma unroll
    for (int j = 0; j < 4; ++j)
      bf[j] = *(const v2f*)(bPtr[j] + k0);
#pragma unroll
    for (int i = 0; i < 2; ++i)
#pragma unroll
      for (int j = 0; j < 4; ++j)
        acc[i][j] = __builtin_amdgcn_wmma_f32_16x16x4_f32(
            /*neg_a=*/false, af[i], /*neg_b=*/false, bf[j],
            /*c_mod=*/(short)0, acc[i][j],
            /*reuse_a=*/false, /*reuse_b=*/false);
  }

  // Epilogue: d = sqrt(max(0, sq[m] + sq[n] - 2*gram)).
  // C/D layout: VGPR r, lane (h*16 + l) holds element (M = r + 8h, N = l).
#pragma unroll
  for (int i = 0; i < 2; ++i) {
#pragma unroll
    for (int j = 0; j < 4; ++j) {
      const int n = n0 + j * 16 + l16;
      const float sqn = sq[n];
#pragma unroll
      for (int r = 0; r < 8; ++r) {
        const int m = m0 + i * 16 + half * 8 + r;
        float d2 = fmaf(-2.0f, acc[i][j][r], sq[m] + sqn);
        d2 = fmaxf(d2, 0.0f);
        out[(size_t)m * EM_N + n] = __builtin_sqrtf(d2);
      }
    }
  }
}

// ---------------------------------------------------------------------------
extern "C" void kernel_launch(void* const* d_in, const int* in_sizes, int n_in,
                              void* d_out, int out_size, void* d_ws, size_t ws_size,
                              hipStream_t stream) {
  (void)in_sizes; (void)n_in; (void)out_size; (void)ws_size;
  const float* A  = (const float*)d_in[0];
  float* out      = (float*)d_out;
  float* sq       = (float*)d_ws;       // 8192 floats = 32 KB scratch

  em_sqnorm<<<EM_N / 8, 256, 0, stream>>>(A, sq);

  dim3 grid(EM_N / 128, EM_N / 128);    // 64 x 64 blocks
  em_dist<<<grid, 256, 0, stream>>>(A, sq, out);
}